// FieldFormer_TAP_53884659695850
// MI455X (gfx1250) — compile-verified
//
#include <hip/hip_runtime.h>
#include <hip/hip_bf16.h>
#include <math.h>

// ---------------------------------------------------------------------------
// FieldFormer TAP on gfx1250: pack -> proj(WMMA) -> BN+L2 -> scores(WMMA)
//   -> sparsemax (+core permute) -> 3x Tucker GEMMs (WMMA) -> tanh
// All GEMM inner loops: two unconditional global_load_b64 + v_wmma_f32_16x16x4.
// f32 WMMA keeps exact dtype parity with the reference (sparsemax support
// count is threshold-sensitive). Workload is latency/L2-bound (~30 MFLOP,
// <1.1 MB working set), so the goal is clean pipelined loops + few launches.
// ---------------------------------------------------------------------------

typedef __attribute__((ext_vector_type(2))) float v2f;
typedef __attribute__((ext_vector_type(8))) float v8f;

#define DIM 125
#define NP  216
#define FD  36
#define RK  20

// Workspace layout (floats)
#define OFF_AP   0         // 224*128 = 28672   padded x
#define OFF_WQ   28672     // 128*128 = 16384   padded qw
#define OFF_WK   45056     // 128*128 = 16384   padded kw
#define OFF_YQ   61440     // 224*128 = 28672   Q (raw -> normalized), padded
#define OFF_YK   90112     // 224*128 = 28672   K
#define OFF_S    118784    // 216*216 = 46656   scores
#define OFF_CP   165440    // 1296*36 = 46656   packed core
#define OFF_T1   212096    // 720*36  = 25920   T1t[(i*20+d)*36 + j]
#define OFF_T2   238016    // 400*36  = 14400   T2t[(c*20+d)*36 + i]
#define OFF_U1   252416    // 32*36 = 1152      padded U1
#define OFF_U2   253568
#define OFF_U3   254720
// total: 255872 floats (~1.0 MB)

// ---------------------------------------------------------------------------
// WMMA f32 16x16x4 accumulate helper. A row-major [M][LDA], B is the
// "n-major" operand: B[k,n] = Bn[n*LDB + k]. K, LDA, LDB compile-time.
// Lane mapping (ISA 7.12.2): A frag a[v] = A[m0+lo][kk + 2*hi + v],
// B frag b[v] = B[kk + 2*hi + v][n0+lo] = Bn[n0+lo][kk + 2*hi + v].
// Both are 8-byte contiguous -> global_load_b64, no guards.
// ---------------------------------------------------------------------------
template <int K, int LDA, int LDB>
__device__ inline v8f wmma_acc_f32(const float* __restrict__ A,
                                   const float* __restrict__ Bn,
                                   int m0, int n0, int lane) {
  int lo = lane & 15, hi = lane >> 4;
  const float* ap = A + (m0 + lo) * LDA + 2 * hi;
  const float* bp = Bn + (n0 + lo) * LDB + 2 * hi;
  v8f c = {};
#pragma unroll
  for (int kk = 0; kk < K; kk += 4) {
    v2f a = *(const v2f*)(ap + kk);
    v2f b = *(const v2f*)(bp + kk);
    c = __builtin_amdgcn_wmma_f32_16x16x4_f32(false, a, false, b, (short)0, c,
                                              false, false);
  }
  return c;
}

// ---------------------------------------------------------------------------
// Kernel 0: pack/pad all GEMM operands; zero the pad rows of Yq/Yk.
// ---------------------------------------------------------------------------
__global__ __launch_bounds__(256) void ff_pack_kernel(
    const float* __restrict__ x, const float* __restrict__ qw,
    const float* __restrict__ kw, const float* __restrict__ U1,
    const float* __restrict__ U2, const float* __restrict__ U3,
    float* __restrict__ ws) {
  int idx = blockIdx.x * 256 + threadIdx.x;
  if (idx < 28672) {  // Ap: 224 x 128
    int m = idx >> 7, k = idx & 127;
    ws[OFF_AP + idx] = (m < NP && k < DIM) ? x[m * DIM + k] : 0.0f;
    return;
  }
  idx -= 28672;
  if (idx < 2 * 16384) {  // Wq/Wk: 128 x 128
    int which = idx >> 14;
    int r = idx & 16383;
    int n = r >> 7, k = r & 127;
    const float* W = which ? kw : qw;
    float v = (n < DIM && k < DIM) ? W[n * DIM + k] : 0.0f;
    ws[(which ? OFF_WK : OFF_WQ) + r] = v;
    return;
  }
  idx -= 2 * 16384;
  if (idx < 3 * 1152) {  // U1/U2/U3: 32 x 36
    int which = idx / 1152;
    int r = idx % 1152;
    int n = r / FD;
    const float* U = (which == 0) ? U1 : (which == 1) ? U2 : U3;
    int off = (which == 0) ? OFF_U1 : (which == 1) ? OFF_U2 : OFF_U3;
    ws[off + r] = (n < RK) ? U[r] : 0.0f;
    return;
  }
  idx -= 3 * 1152;
  if (idx < 2 * 1024) {  // zero pad rows 216..223 of Yq/Yk (8 rows x 128 x 2)
    int which = idx >> 10;
    int r = idx & 1023;
    ws[(which ? OFF_YK : OFF_YQ) + NP * 128 + r] = 0.0f;
  }
}

// ---------------------------------------------------------------------------
// Kernel 1: Y = x @ W^T  into padded Y (stride 128). grid = 14*8*2, block 32.
// ---------------------------------------------------------------------------
__global__ __launch_bounds__(32) void ff_proj_kernel(float* __restrict__ ws) {
  int t = blockIdx.x;
  int which = t / (14 * 8);
  int rem = t % (14 * 8);
  int tm = rem / 8, tn = rem % 8;
  const float* A = ws + OFF_AP;
  const float* W = ws + (which ? OFF_WK : OFF_WQ);
  float* Y = ws + (which ? OFF_YK : OFF_YQ);
  int lane = threadIdx.x, lo = lane & 15, hi = lane >> 4;
  v8f c = wmma_acc_f32<128, 128, 128>(A, W, tm * 16, tn * 16, lane);
#pragma unroll
  for (int v = 0; v < 8; ++v) {
    int m = tm * 16 + v + 8 * hi;
    int n = tn * 16 + lo;
    if (m < NP) Y[m * 128 + n] = c[v];
  }
}

// ---------------------------------------------------------------------------
// Kernel 2: per-row BatchNorm (biased var, train) + L2 normalize, in place on
// padded rows (writes zeros into cols 125..127). grid = 432, block = 128.
// ---------------------------------------------------------------------------
__global__ __launch_bounds__(128) void ff_bn_l2_kernel(
    float* __restrict__ ws, const float* __restrict__ gamma,
    const float* __restrict__ beta) {
  int blk = blockIdx.x;
  int row = blk % NP;
  float* Yrow = ws + (blk / NP ? OFF_YK : OFF_YQ) + row * 128;
  int tid = threadIdx.x;
  __shared__ float s1[128], s2[128];
  __shared__ float smu, srstd, snrm;
  float v = (tid < DIM) ? Yrow[tid] : 0.0f;
  s1[tid] = v;
  s2[tid] = v * v;
  __syncthreads();
  for (int s = 64; s > 0; s >>= 1) {
    if (tid < s) { s1[tid] += s1[tid + s]; s2[tid] += s2[tid + s]; }
    __syncthreads();
  }
  if (tid == 0) {
    float mu = s1[0] * (1.0f / DIM);
    float var = s2[0] * (1.0f / DIM) - mu * mu;
    smu = mu;
    srstd = rsqrtf(var + 1e-5f);
  }
  __syncthreads();
  float yn = (tid < DIM) ? ((v - smu) * srstd * gamma[row] + beta[row]) : 0.0f;
  s1[tid] = yn * yn;
  __syncthreads();
  for (int s = 64; s > 0; s >>= 1) {
    if (tid < s) s1[tid] += s1[tid + s];
    __syncthreads();
  }
  if (tid == 0) snrm = fmaxf(sqrtf(s1[0]), 1e-12f);
  __syncthreads();
  Yrow[tid] = (tid < DIM) ? yn / snrm : 0.0f;
}

// ---------------------------------------------------------------------------
// Kernel 3: S = Qn @ Kn^T / sqrt(125). M=N=224 tile-exact, K=128. No guards
// in loop (pad rows/cols are zero). grid = 14*14, block = 32.
// ---------------------------------------------------------------------------
__global__ __launch_bounds__(32) void ff_scores_kernel(float* __restrict__ ws) {
  int t = blockIdx.x;
  int tm = t / 14, tn = t % 14;
  const float* Q = ws + OFF_YQ;
  const float* Kn = ws + OFF_YK;  // B[k,n] = Kn[n,k]
  float* S = ws + OFF_S;
  int lane = threadIdx.x, lo = lane & 15, hi = lane >> 4;
  v8f c = wmma_acc_f32<128, 128, 128>(Q, Kn, tm * 16, tn * 16, lane);
  const float scale = 0.08944271909999159f;  // 1/sqrt(125)
#pragma unroll
  for (int v = 0; v < 8; ++v) {
    int m = tm * 16 + v + 8 * hi;
    int n = tn * 16 + lo;
    if (m < NP && n < NP) S[m * NP + n] = c[v] * scale;
  }
}

// ---------------------------------------------------------------------------
// Kernel 4: sparsemax per row; writes att_map into d_out+8000 AND scatters the
// value into the packed core Cp[(i*36+j)*36 + k] (the 6^6 transpose).
// grid = 216, block = 256.
// ---------------------------------------------------------------------------
__global__ __launch_bounds__(256) void ff_sparsemax_kernel(
    float* __restrict__ ws, float* __restrict__ att) {
  const float* S = ws + OFF_S;
  float* Cp = ws + OFF_CP;
  __shared__ float z[256];
  __shared__ float cs[256];
  __shared__ float stau;
  __shared__ int scount;
  int row = blockIdx.x, tid = threadIdx.x;
  float val = (tid < NP) ? S[row * NP + tid] : -1e30f;
  z[tid] = val;
  __syncthreads();
  // bitonic sort, descending
  for (int ksz = 2; ksz <= 256; ksz <<= 1) {
    for (int j = ksz >> 1; j > 0; j >>= 1) {
      int ixj = tid ^ j;
      if (ixj > tid) {
        float A = z[tid], B = z[ixj];
        if (((tid & ksz) == 0) == (A < B)) { z[tid] = B; z[ixj] = A; }
      }
      __syncthreads();
    }
  }
  // inclusive cumsum (Hillis-Steele)
  cs[tid] = z[tid];
  __syncthreads();
  for (int off = 1; off < 256; off <<= 1) {
    float tmp = (tid >= off) ? cs[tid - off] : 0.0f;
    __syncthreads();
    cs[tid] += tmp;
    __syncthreads();
  }
  if (tid == 0) scount = 0;
  __syncthreads();
  if (tid < NP) {
    if (1.0f + (float)(tid + 1) * z[tid] > cs[tid]) atomicAdd(&scount, 1);
  }
  __syncthreads();
  if (tid == 0) {
    int kz = max(scount, 1);
    stau = (cs[kz - 1] - 1.0f) / (float)kz;
  }
  __syncthreads();
  if (tid < NP) {
    float r = fmaxf(val - stau, 0.0f);
    att[row * NP + tid] = r;
    // core permute: p=row, q=tid
    int p = row, q = tid;
    int i = 6 * (p / 36) + (q / 36);
    int j = 6 * ((p / 6) % 6) + ((q / 6) % 6);
    int k = 6 * (p % 6) + (q % 6);
    Cp[(i * FD + j) * FD + k] = r;
  }
}

// ---------------------------------------------------------------------------
// Tucker stage A: T1[i,j,d] = sum_k core[i,j,k] U3[d,k].
// A = Cp (1296x36), B = U3p. M=1296 tile-exact. Stores transposed:
// T1t[(i*20+d)*36 + j]. grid = 81*2, block = 32.
// ---------------------------------------------------------------------------
__global__ __launch_bounds__(32) void ff_tuckerA_kernel(float* __restrict__ ws) {
  int t = blockIdx.x;
  int tm = t / 2, tn = t % 2;
  int lane = threadIdx.x, lo = lane & 15, hi = lane >> 4;
  v8f c = wmma_acc_f32<FD, FD, FD>(ws + OFF_CP, ws + OFF_U3, tm * 16, tn * 16,
                                   lane);
  float* T1t = ws + OFF_T1;
#pragma unroll
  for (int v = 0; v < 8; ++v) {
    int r = tm * 16 + v + 8 * hi;  // r = i*36 + j
    int n = tn * 16 + lo;          // n = d
    if (n < RK) T1t[((r / FD) * RK + n) * FD + (r % FD)] = c[v];
  }
}

// ---------------------------------------------------------------------------
// Tucker stage B: T2[i,c,d] = sum_j T1[i,j,d] U2[c,j].
// A = T1t (720x36 contiguous), B = U2p. M=720 tile-exact. Stores transposed:
// T2t[(c*20+d)*36 + i]. grid = 45*2, block = 32.
// ---------------------------------------------------------------------------
__global__ __launch_bounds__(32) void ff_tuckerB_kernel(float* __restrict__ ws) {
  int t = blockIdx.x;
  int tm = t / 2, tn = t % 2;
  int lane = threadIdx.x, lo = lane & 15, hi = lane >> 4;
  v8f c = wmma_acc_f32<FD, FD, FD>(ws + OFF_T1, ws + OFF_U2, tm * 16, tn * 16,
                                   lane);
  float* T2t = ws + OFF_T2;
#pragma unroll
  for (int v = 0; v < 8; ++v) {
    int r = tm * 16 + v + 8 * hi;  // r = i*20 + d
    int n = tn * 16 + lo;          // n = c
    if (n < RK) T2t[(n * RK + (r % RK)) * FD + (r / RK)] = c[v];
  }
}

// ---------------------------------------------------------------------------
// Tucker stage C: out[a,c,d] = tanh( sum_i T2[i,c,d] U1[a,i] ).
// A = T2t (400x36 contiguous), B = U1p. M=400 tile-exact.
// grid = 25*2, block = 32.
// ---------------------------------------------------------------------------
__global__ __launch_bounds__(32) void ff_tuckerC_kernel(float* __restrict__ ws,
                                                        float* __restrict__ out) {
  int t = blockIdx.x;
  int tm = t / 2, tn = t % 2;
  int lane = threadIdx.x, lo = lane & 15, hi = lane >> 4;
  v8f c = wmma_acc_f32<FD, FD, FD>(ws + OFF_T2, ws + OFF_U1, tm * 16, tn * 16,
                                   lane);
#pragma unroll
  for (int v = 0; v < 8; ++v) {
    int r = tm * 16 + v + 8 * hi;  // r = c*20 + d
    int n = tn * 16 + lo;          // n = a
    if (n < RK) out[n * 400 + r] = tanhf(c[v]);
  }
}

// ---------------------------------------------------------------------------
extern "C" void kernel_launch(void* const* d_in, const int* in_sizes, int n_in,
                              void* d_out, int out_size, void* d_ws,
                              size_t ws_size, hipStream_t stream) {
  const float* patches = (const float*)d_in[0];
  const float* qw      = (const float*)d_in[1];
  const float* kw      = (const float*)d_in[2];
  const float* gamma   = (const float*)d_in[3];
  const float* beta    = (const float*)d_in[4];
  const float* U1      = (const float*)d_in[5];
  const float* U2      = (const float*)d_in[6];
  const float* U3      = (const float*)d_in[7];

  float* out = (float*)d_out;       // [0:8000) tanh(tucker), [8000:54656) att
  float* att = out + RK * RK * RK;
  float* ws = (float*)d_ws;

  ff_pack_kernel<<<(66944 + 255) / 256, 256, 0, stream>>>(patches, qw, kw, U1,
                                                          U2, U3, ws);
  ff_proj_kernel<<<14 * 8 * 2, 32, 0, stream>>>(ws);
  ff_bn_l2_kernel<<<2 * NP, 128, 0, stream>>>(ws, gamma, beta);
  ff_scores_kernel<<<14 * 14, 32, 0, stream>>>(ws);
  ff_sparsemax_kernel<<<NP, 256, 0, stream>>>(ws, att);
  ff_tuckerA_kernel<<<81 * 2, 32, 0, stream>>>(ws);
  ff_tuckerB_kernel<<<45 * 2, 32, 0, stream>>>(ws);
  ff_tuckerC_kernel<<<25 * 2, 32, 0, stream>>>(ws, out);
}